// LSA_7524782702974
// MI455X (gfx1250) — compile-verified
//
#include <hip/hip_runtime.h>
#include <stdint.h>

#define BB 16
#define CC 256
#define LL 16384
#define EPSF 1e-6f
#define NTHREADS 256
#define PHASES 2
#define VPP 8                       // float4 vectors per thread per phase
#define ROWS (BB * CC)              // 4096
#define MASK_LEN ((3 * LL) / 4)     // 12288

typedef float v4f __attribute__((ext_vector_type(4)));  // native vector for NT store

// ---------------------------------------------------------------------------
// CDNA5 async global -> LDS copy (GLOBAL_LOAD_ASYNC_TO_LDS_B128, tracked by
// ASYNCcnt). lds_byte_off is the workgroup-relative LDS byte address
// (= low 32 bits of the flat pointer; the shared aperture lives in [63:32]).
// ---------------------------------------------------------------------------
__device__ __forceinline__ void async_copy_b128(uint32_t lds_byte_off, const float* gptr) {
    uint64_t ga = (uint64_t)(uintptr_t)gptr;
    asm volatile("global_load_async_to_lds_b128 %0, %1, off"
                 :: "v"(lds_byte_off), "v"(ga)
                 : "memory");
}

// Async loads complete in order -> waiting ASYNCcnt<=N means the first
// (issued-N) copies have landed in LDS.
__device__ __forceinline__ void wait_async_le4() {
    asm volatile("s_wait_asynccnt 4" ::: "memory");
}
__device__ __forceinline__ void wait_async_zero() {
    asm volatile("s_wait_asynccnt 0" ::: "memory");
}

__device__ __forceinline__ uint32_t lds_offset_of(const void* shared_generic_ptr) {
    return (uint32_t)(uintptr_t)shared_generic_ptr;
}

// ---------------------------------------------------------------------------
// K1: per-(b,c) row mean / std over L (unbiased, +EPS).  One block per row.
// Row streamed through LDS via async B128 copies; each thread consumes only
// the slots it copied -> no barrier between copy and consume, only ASYNCcnt.
// Drain is staged (<=4, then 0) to overlap consumption with in-flight copies.
// ---------------------------------------------------------------------------
__global__ __launch_bounds__(NTHREADS) void k1_row_stats(const float* __restrict__ x,
                                                         float* __restrict__ mean_o,
                                                         float* __restrict__ std_o) {
    __shared__ float4 tile[NTHREADS * VPP];   // 32 KB staging
    __shared__ float red_s[NTHREADS / 32];
    __shared__ float red_ss[NTHREADS / 32];

    const int row = blockIdx.x;
    const int t   = threadIdx.x;
    const float* rowp = x + (size_t)row * LL;
    const uint32_t tile_base = lds_offset_of(&tile[0]);

    float s = 0.0f, ss = 0.0f;

    for (int p = 0; p < PHASES; ++p) {
        const int phase_vec0 = p * NTHREADS * VPP;
        #pragma unroll
        for (int i = 0; i < VPP; ++i) {
            const int v = phase_vec0 + i * NTHREADS + t;        // float4 index in row
            const uint32_t slot = (uint32_t)(i * NTHREADS + t) * 16u;
            async_copy_b128(tile_base + slot, rowp + (size_t)v * 4);
        }
        wait_async_le4();                    // first 4 copies have landed
        #pragma unroll
        for (int i = 0; i < VPP / 2; ++i) {
            float4 d = tile[i * NTHREADS + t];
            s  += (d.x + d.y) + (d.z + d.w);
            ss += d.x * d.x + d.y * d.y + d.z * d.z + d.w * d.w;
        }
        wait_async_zero();                   // remaining 4 have landed
        #pragma unroll
        for (int i = VPP / 2; i < VPP; ++i) {
            float4 d = tile[i * NTHREADS + t];
            s  += (d.x + d.y) + (d.z + d.w);
            ss += d.x * d.x + d.y * d.y + d.z * d.z + d.w * d.w;
        }
    }

    // wave32 reduction
    #pragma unroll
    for (int o = 16; o > 0; o >>= 1) {
        s  += __shfl_xor(s,  o, 32);
        ss += __shfl_xor(ss, o, 32);
    }
    const int wave = t >> 5, lane = t & 31;
    if (lane == 0) { red_s[wave] = s; red_ss[wave] = ss; }
    __syncthreads();
    if (t == 0) {
        float S = 0.0f, SS = 0.0f;
        #pragma unroll
        for (int w = 0; w < NTHREADS / 32; ++w) { S += red_s[w]; SS += red_ss[w]; }
        const float mean = S * (1.0f / (float)LL);
        const float var  = (SS - (float)LL * mean * mean) * (1.0f / (float)(LL - 1));
        mean_o[row] = mean;
        std_o[row]  = sqrtf(var + EPSF);
    }
}

// ---------------------------------------------------------------------------
// K2: cross-batch spread of stats + fused affine params.
//   sqrtvar_mu[c]  = sqrt(var_b(mean[:,c], ddof=1) + EPS)
//   sqrtvar_std[c] = sqrt(var_b(std[:,c],  ddof=1) + EPS)
//   beta  = mean + eps_mu  * sqrtvar_mu ; gamma = std + eps_std * sqrtvar_std
//   out = (x-mean)/std*gamma + beta == x*scale + shift
//   scale = gamma/std ; shift = beta - mean*scale
// ---------------------------------------------------------------------------
__global__ __launch_bounds__(CC) void k2_params(const float* __restrict__ mean_i,
                                                const float* __restrict__ std_i,
                                                const float* __restrict__ eps_mu,
                                                const float* __restrict__ eps_std,
                                                float* __restrict__ scale_o,
                                                float* __restrict__ shift_o) {
    const int c = threadIdx.x;
    float m[BB], sd[BB];
    float msum = 0.0f, ssum = 0.0f;
    #pragma unroll
    for (int b = 0; b < BB; ++b) {
        m[b]  = mean_i[b * CC + c];
        sd[b] = std_i[b * CC + c];
        msum += m[b];
        ssum += sd[b];
    }
    const float mbar = msum * (1.0f / (float)BB);
    const float sbar = ssum * (1.0f / (float)BB);
    float mv = 0.0f, sv = 0.0f;
    #pragma unroll
    for (int b = 0; b < BB; ++b) {
        const float dm = m[b] - mbar;
        const float ds = sd[b] - sbar;
        mv += dm * dm;
        sv += ds * ds;
    }
    const float sq_mu = sqrtf(mv * (1.0f / (float)(BB - 1)) + EPSF);
    const float sq_sd = sqrtf(sv * (1.0f / (float)(BB - 1)) + EPSF);
    #pragma unroll
    for (int b = 0; b < BB; ++b) {
        const float beta  = m[b]  + eps_mu[b * CC + c]  * sq_mu;
        const float gamma = sd[b] + eps_std[b * CC + c] * sq_sd;
        const float sc    = gamma / sd[b];
        scale_o[b * CC + c] = sc;
        shift_o[b * CC + c] = beta - m[b] * sc;
    }
}

// ---------------------------------------------------------------------------
// K3: fused normalize + temporal mask.  One block per row; async-staged
// reads, single FMA in the mask window, non-temporal B128 stores (output is
// write-once, never re-read -> keep it out of L2's way).
// ---------------------------------------------------------------------------
__global__ __launch_bounds__(NTHREADS) void k3_apply(const float* __restrict__ x,
                                                     const float* __restrict__ scale_i,
                                                     const float* __restrict__ shift_i,
                                                     const int* __restrict__ starts,
                                                     float* __restrict__ out) {
    __shared__ float4 tile[NTHREADS * VPP];   // 32 KB staging

    const int row = blockIdx.x;               // b*CC + c
    const int b   = row / CC;
    const int t   = threadIdx.x;
    const float sc = scale_i[row];
    const float sh = shift_i[row];
    const int st = starts[b];
    const int en = st + MASK_LEN;

    const float* rowp = x + (size_t)row * LL;
    float*       outp = out + (size_t)row * LL;
    const uint32_t tile_base = lds_offset_of(&tile[0]);

    for (int p = 0; p < PHASES; ++p) {
        const int phase_vec0 = p * NTHREADS * VPP;
        #pragma unroll
        for (int i = 0; i < VPP; ++i) {
            const int v = phase_vec0 + i * NTHREADS + t;
            const uint32_t slot = (uint32_t)(i * NTHREADS + t) * 16u;
            async_copy_b128(tile_base + slot, rowp + (size_t)v * 4);
        }
        wait_async_le4();
        #pragma unroll
        for (int i = 0; i < VPP; ++i) {
            if (i == VPP / 2) wait_async_zero();
            const int v = phase_vec0 + i * NTHREADS + t;
            const float4 d = tile[i * NTHREADS + t];
            const int l = v * 4;
            v4f r;
            r.x = (l + 0 >= st && l + 0 < en) ? fmaf(d.x, sc, sh) : d.x;
            r.y = (l + 1 >= st && l + 1 < en) ? fmaf(d.y, sc, sh) : d.y;
            r.z = (l + 2 >= st && l + 2 < en) ? fmaf(d.z, sc, sh) : d.z;
            r.w = (l + 3 >= st && l + 3 < en) ? fmaf(d.w, sc, sh) : d.w;
            __builtin_nontemporal_store(r, (v4f*)(outp + l));
        }
    }
}

// ---------------------------------------------------------------------------
extern "C" void kernel_launch(void* const* d_in, const int* in_sizes, int n_in,
                              void* d_out, int out_size, void* d_ws, size_t ws_size,
                              hipStream_t stream) {
    const float* x       = (const float*)d_in[0];   // [B,C,L]
    const float* eps_mu  = (const float*)d_in[1];   // [B,C]
    const float* eps_std = (const float*)d_in[2];   // [B,C]
    const int*   starts  = (const int*)d_in[3];     // [B]
    float* out = (float*)d_out;

    float* ws    = (float*)d_ws;
    float* meanw = ws;                // ROWS floats
    float* stdw  = ws + ROWS;         // ROWS floats
    float* scale = ws + 2 * ROWS;     // ROWS floats
    float* shift = ws + 3 * ROWS;     // ROWS floats

    k1_row_stats<<<ROWS, NTHREADS, 0, stream>>>(x, meanw, stdw);
    k2_params<<<1, CC, 0, stream>>>(meanw, stdw, eps_mu, eps_std, scale, shift);
    k3_apply<<<ROWS, NTHREADS, 0, stream>>>(x, scale, shift, starts, out);
}